// Net_82995948028407
// MI455X (gfx1250) — compile-verified
//
#include <hip/hip_runtime.h>
#include <hip/hip_bf16.h>
#include <math.h>

// Problem constants (match reference)
#define NFEAT   512
#define HIDDEN  16
#define NCLS    16
#define EPSN    1e-12f

typedef __attribute__((ext_vector_type(2))) float v2f;
typedef __attribute__((ext_vector_type(8))) float v8f;

// ---- monotonic float<->uint encoding so unsigned atomicMax == float max ----
__device__ __forceinline__ unsigned fenc(float f) {
  unsigned u = __float_as_uint(f);
  return (u & 0x80000000u) ? ~u : (u | 0x80000000u);
}
__device__ __forceinline__ float fdec(unsigned u) {
  return __uint_as_float((u & 0x80000000u) ? (u & 0x7fffffffu) : ~u);
}

// ============================================================================
// Kernel 1: h = relu(x @ W1 + b1), fp32 WMMA 16x16x4
// One wave -> one 16x16 output tile, K = 512 -> 128 WMMA steps.
// W1 (512x16 = 32KB) staged in LDS, shared by all 8 waves of the block.
// A stream (205 MB, read once) pulled ahead with global_prefetch_b8.
// ============================================================================
__global__ __launch_bounds__(256) void gemm1_relu_wmma(
    const float* __restrict__ x, const float* __restrict__ W1,
    const float* __restrict__ b1, float* __restrict__ h, int n)
{
  __shared__ float sB[NFEAT * HIDDEN]; // 32 KB
  {
    const float4* W4  = (const float4*)W1;
    float4*       sB4 = (float4*)sB;
    for (int i = threadIdx.x; i < (NFEAT * HIDDEN) / 4; i += 256)
      sB4[i] = W4[i];
  }
  __syncthreads();

  const int wave   = threadIdx.x >> 5;
  const int lane   = threadIdx.x & 31;
  const int tile   = blockIdx.x * 8 + wave;
  const int nTiles = (n + 15) >> 4;
  if (tile >= nTiles) return;          // wave-uniform: EXEC stays all-ones

  const int  m  = lane & 15;           // A row / B col / D col for this lane
  const int  kk = (lane >> 4) << 1;    // K sub-offset: 0 (lanes 0-15) or 2
  long row = (long)tile * 16 + m;
  if (row >= n) row = n - 1;           // clamp loads (n is a multiple of 16 here)
  const float* arow = x + row * NFEAT;

  v8f acc = {};
  for (int kc = 0; kc < NFEAT; kc += 16) {
    // pull the next cachelines of this lane's A row into L0/L2 ahead of use
    if (kc + 128 < NFEAT)
      __builtin_prefetch(arow + kc + 128, 0, 3);
    #pragma unroll
    for (int k0 = kc; k0 < kc + 16; k0 += 4) {
      // A fragment: 16x4 f32; lane holds (m, k0+kk) and (m, k0+kk+1)
      const float2 av = *(const float2*)(arow + k0 + kk);
      v2f a; a.x = av.x; a.y = av.y;
      // B fragment: 4x16 f32; lane holds (k0+kk, m) and (k0+kk+1, m)
      v2f b;
      b.x = sB[(k0 + kk)     * HIDDEN + m];
      b.y = sB[(k0 + kk + 1) * HIDDEN + m];
      acc = __builtin_amdgcn_wmma_f32_16x16x4_f32(
          /*neg_a=*/false, a, /*neg_b=*/false, b,
          /*c_mod=*/(short)0, acc, /*reuse_a=*/false, /*reuse_b=*/false);
    }
  }

  // D layout: VGPR j -> row (tile*16 + j + 8*(lane>=16)), col m
  const float bias  = b1[m];
  const int   rhalf = (lane >> 4) << 3;
  float* hp = h + ((long)tile * 16 + rhalf) * HIDDEN + m;
  const bool fullTile = ((long)tile * 16 + 16) <= (long)n;  // uniform per wave
  if (fullTile) {
    #pragma unroll
    for (int j = 0; j < 8; ++j) {
      float v = acc[j] + bias;
      hp[j * HIDDEN] = v > 0.0f ? v : 0.0f;   // 8 stores, immediate offsets
    }
  } else {
    #pragma unroll
    for (int j = 0; j < 8; ++j) {
      long r = (long)tile * 16 + j + rhalf;
      if (r < n) {
        float v = acc[j] + bias;
        h[r * HIDDEN + m] = v > 0.0f ? v : 0.0f;
      }
    }
  }
}

// ============================================================================
// Row L2-normalize: hn = h / max(||h||, eps)        (one thread per node)
// ============================================================================
__global__ __launch_bounds__(256) void row_normalize(
    const float* __restrict__ h, float* __restrict__ hn, int n)
{
  int i = blockIdx.x * blockDim.x + threadIdx.x;
  if (i >= n) return;
  const float4* hr = (const float4*)(h + (long)i * HIDDEN);
  float4 a = hr[0], b = hr[1], c = hr[2], d = hr[3];
  float s = a.x*a.x + a.y*a.y + a.z*a.z + a.w*a.w
          + b.x*b.x + b.y*b.y + b.z*b.z + b.w*b.w
          + c.x*c.x + c.y*c.y + c.z*c.z + c.w*c.w
          + d.x*d.x + d.y*d.y + d.z*d.z + d.w*d.w;
  float inv = 1.0f / fmaxf(sqrtf(s), EPSN);
  float4* o = (float4*)(hn + (long)i * HIDDEN);
  a.x*=inv; a.y*=inv; a.z*=inv; a.w*=inv;
  b.x*=inv; b.y*=inv; b.z*=inv; b.w*=inv;
  c.x*=inv; c.y*=inv; c.z*=inv; c.w*=inv;
  d.x*=inv; d.y*=inv; d.z*=inv; d.w*=inv;
  o[0]=a; o[1]=b; o[2]=c; o[3]=d;
}

// ============================================================================
// Per-prop init: max = 0 (below enc of any finite float), sum = 0, accum = 0
// (d_ws is poisoned by the harness, so everything we read must be written)
// ============================================================================
__global__ __launch_bounds__(256) void prop_init(
    unsigned* __restrict__ nodeMax, float* __restrict__ nodeSum,
    float* __restrict__ accum, int n)
{
  int i = blockIdx.x * blockDim.x + threadIdx.x;
  if (i >= n) return;
  nodeMax[i] = 0u;
  nodeSum[i] = 0.0f;
  float4 z = {0.f, 0.f, 0.f, 0.f};
  float4* o = (float4*)(accum + (long)i * HIDDEN);
  o[0] = z; o[1] = z; o[2] = z; o[3] = z;
}

// ============================================================================
// Edge pass 1: logits[e] = beta * <hn[src], hn[dst]>; segment-max into nodeMax
// Edges [0,nE) come from edge_index; edges [nE, nE+n) are self-loops.
// ============================================================================
__global__ __launch_bounds__(256) void edge_logit_max(
    const float* __restrict__ hn, const int* __restrict__ esrc,
    const int* __restrict__ edst, const float* __restrict__ betaPtr,
    float* __restrict__ logits, unsigned* __restrict__ nodeMax,
    int nE, int n)
{
  long e = (long)blockIdx.x * blockDim.x + threadIdx.x;
  if (e >= (long)nE + n) return;
  int s, d;
  if (e < nE) { s = esrc[e]; d = edst[e]; } else { s = d = (int)(e - nE); }
  const float4* a = (const float4*)(hn + (long)s * HIDDEN);
  const float4* b = (const float4*)(hn + (long)d * HIDDEN);
  float dot = 0.0f;
  #pragma unroll
  for (int j = 0; j < 4; ++j) {
    float4 av = a[j], bv = b[j];
    dot += av.x*bv.x + av.y*bv.y + av.z*bv.z + av.w*bv.w;
  }
  float beta = betaPtr ? betaPtr[0] : 1.0f;
  float l = beta * dot;
  logits[e] = l;
  atomicMax(&nodeMax[d], fenc(l));
}

// ============================================================================
// Edge pass 2: w = exp(l - max[dst]); logits[e] <- w; segment-sum into nodeSum
// ============================================================================
__global__ __launch_bounds__(256) void edge_exp_sum(
    const int* __restrict__ edst, const unsigned* __restrict__ nodeMax,
    float* __restrict__ logits, float* __restrict__ nodeSum, int nE, int n)
{
  long e = (long)blockIdx.x * blockDim.x + threadIdx.x;
  if (e >= (long)nE + n) return;
  int d = (e < nE) ? edst[e] : (int)(e - nE);
  float w = __expf(logits[e] - fdec(nodeMax[d]));
  logits[e] = w;
  atomicAdd(&nodeSum[d], w);
}

// ============================================================================
// Edge pass 3: alpha = w / sum[dst]; accum[dst] += alpha * h[src]
// ============================================================================
__global__ __launch_bounds__(256) void edge_scatter(
    const float* __restrict__ h, const int* __restrict__ esrc,
    const int* __restrict__ edst, const float* __restrict__ logits,
    const float* __restrict__ nodeSum, float* __restrict__ accum,
    int nE, int n)
{
  long e = (long)blockIdx.x * blockDim.x + threadIdx.x;
  if (e >= (long)nE + n) return;
  int s, d;
  if (e < nE) { s = esrc[e]; d = edst[e]; } else { s = d = (int)(e - nE); }
  float alpha = logits[e] / nodeSum[d];
  const float4* hs = (const float4*)(h + (long)s * HIDDEN);
  float* od = accum + (long)d * HIDDEN;
  #pragma unroll
  for (int j = 0; j < 4; ++j) {
    float4 v = hs[j];
    atomicAdd(od + 4*j + 0, alpha * v.x);
    atomicAdd(od + 4*j + 1, alpha * v.y);
    atomicAdd(od + 4*j + 2, alpha * v.z);
    atomicAdd(od + 4*j + 3, alpha * v.w);
  }
}

// ============================================================================
// Head: out = log_softmax(h @ W2 + b2)              (one thread per node)
// ============================================================================
__global__ __launch_bounds__(256) void head_logsoftmax(
    const float* __restrict__ h, const float* __restrict__ W2,
    const float* __restrict__ b2, float* __restrict__ out, int n)
{
  __shared__ float sW[HIDDEN * NCLS];
  __shared__ float sb[NCLS];
  if (threadIdx.x < HIDDEN * NCLS) sW[threadIdx.x] = W2[threadIdx.x];
  if (threadIdx.x < NCLS)          sb[threadIdx.x] = b2[threadIdx.x];
  __syncthreads();

  int i = blockIdx.x * blockDim.x + threadIdx.x;
  if (i >= n) return;
  float hr[HIDDEN];
  {
    const float4* hp = (const float4*)(h + (long)i * HIDDEN);
    #pragma unroll
    for (int j = 0; j < 4; ++j) {
      float4 v = hp[j];
      hr[4*j+0] = v.x; hr[4*j+1] = v.y; hr[4*j+2] = v.z; hr[4*j+3] = v.w;
    }
  }
  float o[NCLS];
  #pragma unroll
  for (int c = 0; c < NCLS; ++c) o[c] = sb[c];
  #pragma unroll
  for (int k = 0; k < HIDDEN; ++k) {
    float hv = hr[k];
    #pragma unroll
    for (int c = 0; c < NCLS; ++c) o[c] = fmaf(hv, sW[k * NCLS + c], o[c]);
  }
  float mx = o[0];
  #pragma unroll
  for (int c = 1; c < NCLS; ++c) mx = fmaxf(mx, o[c]);
  float se = 0.0f;
  #pragma unroll
  for (int c = 0; c < NCLS; ++c) se += __expf(o[c] - mx);
  float lse = mx + __logf(se);
  float* op = out + (long)i * NCLS;
  #pragma unroll
  for (int c = 0; c < NCLS; ++c) op[c] = o[c] - lse;
}

// ============================================================================
// Host-side orchestration
// ============================================================================
extern "C" void kernel_launch(void* const* d_in, const int* in_sizes, int n_in,
                              void* d_out, int out_size, void* d_ws, size_t ws_size,
                              hipStream_t stream) {
  const float* x     = (const float*)d_in[0];
  const int*   eidx  = (const int*)  d_in[1];   // [2, E] flattened
  const float* W1    = (const float*)d_in[2];
  const float* b1    = (const float*)d_in[3];
  const float* W2    = (const float*)d_in[4];
  const float* b2    = (const float*)d_in[5];
  const float* beta2 = (const float*)d_in[6];

  const int n  = in_sizes[0] / NFEAT;   // 100000
  const int nE = in_sizes[1] / 2;       // 3200000
  const int*  esrc = eidx;
  const int*  edst = eidx + nE;
  const long  eTot = (long)nE + n;

  // Workspace layout (~40 MB total)
  char*  ws      = (char*)d_ws;
  size_t featB   = (size_t)n * HIDDEN * sizeof(float);
  float* h1      = (float*)ws;                 ws += featB;   // relu(xW1+b1)
  float* hn      = (float*)ws;                 ws += featB;   // normalized rows
  float* h2      = (float*)ws;                 ws += featB;   // prop1 output
  float* h3      = (float*)ws;                 ws += featB;   // prop2 output
  unsigned* nMax = (unsigned*)ws;              ws += (size_t)n * sizeof(unsigned);
  float* nSum    = (float*)ws;                 ws += (size_t)n * sizeof(float);
  float* logits  = (float*)ws;                 /* eTot floats */

  const int TB = 256;
  dim3 blkN((n + TB - 1) / TB);
  dim3 blkE((int)((eTot + TB - 1) / TB));

  // ---- layer 1 GEMM (WMMA) ----
  {
    int nTiles = (n + 15) >> 4;
    dim3 grid((nTiles + 7) / 8);
    gemm1_relu_wmma<<<grid, 256, 0, stream>>>(x, W1, b1, h1, n);
  }

  // ---- two AGNN propagation rounds ----
  auto prop = [&](const float* hin, float* hout, const float* betaPtr) {
    row_normalize <<<blkN, TB, 0, stream>>>(hin, hn, n);
    prop_init     <<<blkN, TB, 0, stream>>>(nMax, nSum, hout, n);
    edge_logit_max<<<blkE, TB, 0, stream>>>(hn, esrc, edst, betaPtr, logits, nMax, nE, n);
    edge_exp_sum  <<<blkE, TB, 0, stream>>>(edst, nMax, logits, nSum, nE, n);
    edge_scatter  <<<blkE, TB, 0, stream>>>(hin, esrc, edst, logits, nSum, hout, nE, n);
  };
  prop(h1, h2, nullptr);   // prop1: fixed beta = 1
  prop(h2, h3, beta2);     // prop2: learnable beta

  // ---- head + log_softmax ----
  head_logsoftmax<<<blkN, TB, 0, stream>>>(h3, W2, b2, (float*)d_out, n);
}